// External_attention_81776177315885
// MI455X (gfx1250) — compile-verified
//
#include <hip/hip_runtime.h>
#include <hip/hip_bf16.h>

// External attention, algebraically folded:
//   logits = (l0 @ conv_w) @ x + (l0 @ conv_b)        (k1, WMMA f32 16x16x4)
//   softmax over n, L1 over k                          (k2, k3 — 16MiB, L2-resident)
//   out    = (l1 @ g) * invdenom + x                   (k4, WMMA f32 16x16x4)
// Memory-bound: ~400MB @ 23.3 TB/s ≈ 17us; fp32 WMMA keeps full precision for free.

typedef __attribute__((ext_vector_type(2))) float v2f;
typedef __attribute__((ext_vector_type(8))) float v8f;

#define C_DIM 512
#define K_DIM 64
#define B_DIM 4
#define N_DIM 16384  // 128*128

__device__ __forceinline__ v8f wmma_f32_16x16x4(v2f a, v2f b, v8f c) {
  // 8 args: (neg_a, A, neg_b, B, c_mod, C, reuse_a, reuse_b)
  return __builtin_amdgcn_wmma_f32_16x16x4_f32(false, a, false, b, (short)0, c,
                                               false, false);
}

// ---------------- k0: fold weights: w0f[k][c] = sum_o l0[k][o]*cw[o][c] -------
__global__ void k0_fold(const float* __restrict__ l0, const float* __restrict__ cw,
                        const float* __restrict__ cb, float* __restrict__ w0f,
                        float* __restrict__ b0f) {
  int k = blockIdx.x;  // 64 blocks
  const float* l0r = l0 + k * C_DIM;
  for (int c = threadIdx.x; c < C_DIM; c += blockDim.x) {
    float acc = 0.f;
    for (int o = 0; o < C_DIM; ++o) acc = fmaf(l0r[o], cw[o * C_DIM + c], acc);
    w0f[k * C_DIM + c] = acc;
  }
  if (threadIdx.x == 0) {
    float acc = 0.f;
    for (int o = 0; o < C_DIM; ++o) acc = fmaf(l0r[o], cb[o], acc);
    b0f[k] = acc;
  }
}

// ---------------- k1: logits[b][k][n] = w0f @ x + b0f (WMMA) ------------------
// One wave per 16-wide n-strip; 4 accumulators cover all K=64 rows, so x is
// streamed from HBM exactly once. w0f (128KB) stays hot in L1/L2.
__global__ void k1_logits(const float* __restrict__ x, const float* __restrict__ w0f,
                          const float* __restrict__ b0f, float* __restrict__ logits) {
  int wave = blockIdx.x * (blockDim.x >> 5) + (threadIdx.x >> 5);
  int lane = threadIdx.x & 31;
  int half = lane >> 4;    // 0: K 0..1, 1: K 2..3 (A/B layout, ISA 7.12.2)
  int m    = lane & 15;
  int b     = wave >> 10;             // 1024 strips per batch image
  int nbase = (wave & 1023) << 4;
  const float* xb = x + (size_t)b * C_DIM * N_DIM;

  v8f acc[4];
  v8f zero = {0.f, 0.f, 0.f, 0.f, 0.f, 0.f, 0.f, 0.f};
#pragma unroll
  for (int t = 0; t < 4; ++t) acc[t] = zero;

  for (int c = 0; c < C_DIM; c += 4) {
    int kr = c + half * 2;
    v2f bm;  // B 4x16 tile of x: rows kr,kr+1 for this half-wave
    bm.x = xb[(size_t)kr * N_DIM + nbase + m];
    bm.y = xb[(size_t)(kr + 1) * N_DIM + nbase + m];
#pragma unroll
    for (int t = 0; t < 4; ++t) {
      int arow = t * 16 + m;  // A 16x4 tile of w0f
      v2f am;
      am.x = w0f[arow * C_DIM + kr];
      am.y = w0f[arow * C_DIM + kr + 1];
      acc[t] = wmma_f32_16x16x4(am, bm, acc[t]);
    }
  }

  float* lb = logits + (size_t)b * K_DIM * N_DIM;
#pragma unroll
  for (int t = 0; t < 4; ++t) {
#pragma unroll
    for (int r = 0; r < 8; ++r) {
      int k = t * 16 + r + half * 8;  // C/D layout: VGPR r -> rows r / r+8
      lb[(size_t)k * N_DIM + nbase + m] = acc[t][r] + b0f[k];
    }
  }
}

// ---------------- k2: per-(b,k) row max and 1/sum(exp) over n -----------------
__global__ void k2_rowstats(const float* __restrict__ logits,
                            float* __restrict__ rowmax, float* __restrict__ rowinv) {
  __shared__ float red[256];
  int r = blockIdx.x;  // 256 rows = B*K
  int tid = threadIdx.x;
  const float* p = logits + (size_t)r * N_DIM;
  float mx = -3.4e38f;
  for (int i = tid; i < N_DIM; i += 256) mx = fmaxf(mx, p[i]);
  red[tid] = mx; __syncthreads();
  for (int s = 128; s > 0; s >>= 1) {
    if (tid < s) red[tid] = fmaxf(red[tid], red[tid + s]);
    __syncthreads();
  }
  mx = red[0]; __syncthreads();
  float sum = 0.f;
  for (int i = tid; i < N_DIM; i += 256) sum += __expf(p[i] - mx);
  red[tid] = sum; __syncthreads();
  for (int s = 128; s > 0; s >>= 1) {
    if (tid < s) red[tid] += red[tid + s];
    __syncthreads();
  }
  if (tid == 0) { rowmax[r] = mx; rowinv[r] = 1.f / red[0]; }
}

// ---------------- k3: g = exp(l-max)/S in place; invdenom over k --------------
__global__ void k3_norm(float* __restrict__ logits, const float* __restrict__ rowmax,
                        const float* __restrict__ rowinv, float* __restrict__ invden) {
  int idx = blockIdx.x * blockDim.x + threadIdx.x;  // over B*N
  int b = idx >> 14;
  int n = idx & (N_DIM - 1);
  float* lb = logits + (size_t)b * K_DIM * N_DIM + n;
  const float* mxp = rowmax + b * K_DIM;
  const float* isp = rowinv + b * K_DIM;
  float acc = 0.f;
#pragma unroll 4
  for (int k = 0; k < K_DIM; ++k) {
    float g = __expf(lb[(size_t)k * N_DIM] - mxp[k]) * isp[k];
    lb[(size_t)k * N_DIM] = g;
    acc += g;
  }
  invden[idx] = 1.f / (1e-9f + acc);
}

// ---------------- k4: out = (l1 @ g) * invdenom + x (WMMA) --------------------
// One block per 16-wide n-strip; 8 waves x 4 c-tiles cover C=512.
__global__ void k4_out(const float* __restrict__ x, const float* __restrict__ g,
                       const float* __restrict__ l1, const float* __restrict__ invden,
                       float* __restrict__ out) {
  int w    = threadIdx.x >> 5;  // wave 0..7
  int lane = threadIdx.x & 31;
  int half = lane >> 4;
  int m    = lane & 15;
  int strip = blockIdx.x;
  int b     = strip >> 10;
  int nbase = (strip & 1023) << 4;
  const float* gb = g + (size_t)b * K_DIM * N_DIM;

  v8f acc[4];
  v8f zero = {0.f, 0.f, 0.f, 0.f, 0.f, 0.f, 0.f, 0.f};
#pragma unroll
  for (int j = 0; j < 4; ++j) acc[j] = zero;

#pragma unroll
  for (int kk = 0; kk < K_DIM; kk += 4) {
    int kr = kk + half * 2;
    v2f bm;
    bm.x = gb[(size_t)kr * N_DIM + nbase + m];
    bm.y = gb[(size_t)(kr + 1) * N_DIM + nbase + m];
#pragma unroll
    for (int j = 0; j < 4; ++j) {
      int crow = (w * 4 + j) * 16 + m;  // A tile of l1 [C,K]
      v2f am;
      am.x = l1[crow * K_DIM + kr];
      am.y = l1[crow * K_DIM + kr + 1];
      acc[j] = wmma_f32_16x16x4(am, bm, acc[j]);
    }
  }

  float id = invden[(b << 14) + nbase + m];
  const float* xb = x + (size_t)b * C_DIM * N_DIM;
  float* ob = out + (size_t)b * C_DIM * N_DIM;
#pragma unroll
  for (int j = 0; j < 4; ++j) {
#pragma unroll
    for (int r = 0; r < 8; ++r) {
      int c = (w * 4 + j) * 16 + r + half * 8;
      size_t off = (size_t)c * N_DIM + nbase + m;
      ob[off] = fmaf(acc[j][r], id, xb[off]);
    }
  }
}

extern "C" void kernel_launch(void* const* d_in, const int* in_sizes, int n_in,
                              void* d_out, int out_size, void* d_ws, size_t ws_size,
                              hipStream_t stream) {
  const float* x  = (const float*)d_in[0];  // [B,C,H,W]
  const float* cw = (const float*)d_in[1];  // [C,C]
  const float* cb = (const float*)d_in[2];  // [C]
  const float* l0 = (const float*)d_in[3];  // [K,C]
  const float* l1 = (const float*)d_in[4];  // [C,K]
  float* out = (float*)d_out;

  char* ws = (char*)d_ws;
  float* w0f    = (float*)(ws);              // 64*512 f32 = 128KB
  float* b0f    = (float*)(ws + 131072);     // 64 f32
  float* rowmax = (float*)(ws + 132096);     // 256 f32
  float* rowinv = (float*)(ws + 133632);     // 256 f32
  float* invden = (float*)(ws + 135168);     // 65536 f32 = 256KB
  float* logits = (float*)(ws + (1u << 20)); // 16 MiB [B,K,N] (reused as g)

  k0_fold    <<<K_DIM, 256, 0, stream>>>(l0, cw, cb, w0f, b0f);
  k1_logits  <<<1024, 128, 0, stream>>>(x, w0f, b0f, logits);   // 4096 waves
  k2_rowstats<<<B_DIM * K_DIM, 256, 0, stream>>>(logits, rowmax, rowinv);
  k3_norm    <<<(B_DIM * N_DIM) / 256, 256, 0, stream>>>(logits, rowmax, rowinv, invden);
  k4_out     <<<B_DIM * (N_DIM / 16), 256, 0, stream>>>(x, logits, l1, invden, out);
}